// DeepRichardsonLucy_79886391705891
// MI455X (gfx1250) — compile-verified
//
#include <hip/hip_runtime.h>

typedef float v2f __attribute__((ext_vector_type(2)));
typedef float v8f __attribute__((ext_vector_type(8)));

#define BATCH 4096
#define LEN   8192
#define KTAP  31
#define CHUNK 256          // output positions per block
#define ROWS  16           // batch rows per block (WMMA M)
#define NTH   128          // 4 waves of 32
#define NWAVE 4
#define SW    (CHUNK + 64) // s tile payload width  (halo 32 each side, 16-aligned)
#define RW    (CHUNK + 32) // ratio/x tile payload width (halo 16 each side)
#define SSTR  (SW + 1)     // padded LDS row strides (odd multiple -> conflict-free)
#define RSTR  (RW + 1)
#define T1    (RW / 16)    // 18 conv1 output tiles (covers ratio region)
#define T2    (CHUNK / 16) // 16 conv2 output tiles
#define KC    12           // 48 = 15-halo + 16 outputs + 15-halo + pad, in K=4 chunks
#define EPSV  1e-6f

// fast reciprocal: v_rcp_f32 + one Newton-Raphson step (~full fp32 accuracy,
// 3 VALU ops vs ~12 for the IEEE div_scale/div_fmas sequence)
__device__ __forceinline__ float fast_rcp(float d) {
    float r = __builtin_amdgcn_rcpf(d);
    r = r * (2.0f - d * r);
    return r;
}

__global__ __launch_bounds__(NTH)
void rl_layer_kernel(const float* __restrict__ x,
                     const float* __restrict__ psf,
                     const float* __restrict__ alpha,
                     const float* __restrict__ s_in,
                     float* __restrict__ s_out,
                     int layer, int first)
{
    __shared__ float s_lds[ROWS * SSTR];   // s tile, origin at pos c0-32
    __shared__ float r_lds[ROWS * RSTR];   // ratio tile, origin at pos c0-16
    __shared__ float x_lds[ROWS * RSTR];   // x tile,     origin at pos c0-16
    __shared__ float psf_s[KTAP];

    const int tid  = threadIdx.x;
    const int lane = tid & 31;
    const int wave = tid >> 5;
    const int hi   = lane >> 4;   // half-wave select
    const int n    = lane & 15;   // WMMA N (column) / A-matrix M row

    const int c0 = blockIdx.x * CHUNK;
    const int r0 = blockIdx.y * ROWS;

    if (tid < KTAP) psf_s[tid] = psf[tid];

    // ---- cooperative tile loads (zero-pad outside [0,LEN)) ----
    for (int i = tid; i < ROWS * SW; i += NTH) {
        int m = i / SW, j = i - m * SW;
        int pos = c0 - 32 + j;
        float v = 0.0f;
        if (pos >= 0 && pos < LEN)
            v = first ? 0.5f : s_in[(size_t)(r0 + m) * LEN + pos];
        s_lds[m * SSTR + j] = v;
    }
    for (int i = tid; i < ROWS * RW; i += NTH) {
        int m = i / RW, j = i - m * RW;
        int pos = c0 - 16 + j;
        float v = 0.0f;
        if (pos >= 0 && pos < LEN)
            v = x[(size_t)(r0 + m) * LEN + pos];
        x_lds[m * RSTR + j] = v;
    }
    __syncthreads();

    // ---- per-lane Toeplitz B operands (ISA 4x16 B layout: N=lane&15,
    //      K = kc*4 + 2*hi + {0,1});  B[k][n] = w[k-n-1] ----
    v2f Bw1[KC], Bw2[KC];
    #pragma unroll
    for (int kc = 0; kc < KC; ++kc) {
        #pragma unroll
        for (int d = 0; d < 2; ++d) {
            int k = kc * 4 + hi * 2 + d;
            int t = k - n - 1;
            float w1 = 0.0f, w2 = 0.0f;
            if (t >= 0 && t < KTAP) { w1 = psf_s[t]; w2 = psf_s[KTAP - 1 - t]; }
            Bw1[kc][d] = w1;
            Bw2[kc][d] = w2;
        }
    }

    // ---- conv1 (s * psf) over ratio region, then ratio = x / (conv+eps) ----
    for (int t = wave; t < T1; t += NWAVE) {
        v8f acc = {};
        const float* sa = &s_lds[n * SSTR + t * 16 + hi * 2];
        #pragma unroll
        for (int kc = 0; kc < KC; ++kc) {
            v2f a;
            a.x = sa[kc * 4 + 0];
            a.y = sa[kc * 4 + 1];
            acc = __builtin_amdgcn_wmma_f32_16x16x4_f32(
                false, a, false, Bw1[kc], (short)0, acc, false, false);
        }
        #pragma unroll
        for (int v = 0; v < 8; ++v) {
            int m = v + 8 * hi;                 // C/D layout: M = v + 8*hi, N = n
            int idx = t * 16 + n;
            r_lds[m * RSTR + idx] = x_lds[m * RSTR + idx] * fast_rcp(acc[v] + EPSV);
        }
    }
    __syncthreads();

    const float a_l = alpha[layer];

    // ---- conv2 (ratio * psf_flip), then s = max(s * corr^alpha, eps) ----
    for (int t = wave; t < T2; t += NWAVE) {
        v8f acc = {};
        const float* ra = &r_lds[n * RSTR + t * 16 + hi * 2];
        #pragma unroll
        for (int kc = 0; kc < KC; ++kc) {
            v2f a;
            a.x = ra[kc * 4 + 0];
            a.y = ra[kc * 4 + 1];
            acc = __builtin_amdgcn_wmma_f32_16x16x4_f32(
                false, a, false, Bw2[kc], (short)0, acc, false, false);
        }
        #pragma unroll
        for (int v = 0; v < 8; ++v) {
            int m = v + 8 * hi;
            float corr = fmaxf(acc[v] + EPSV, EPSV);
            float sold = s_lds[m * SSTR + 32 + t * 16 + n];
            float sn   = fmaxf(sold * __powf(corr, a_l), EPSV);
            s_out[(size_t)(r0 + m) * LEN + c0 + t * 16 + n] = sn;
        }
    }
}

extern "C" void kernel_launch(void* const* d_in, const int* in_sizes, int n_in,
                              void* d_out, int out_size, void* d_ws, size_t ws_size,
                              hipStream_t stream) {
    const float* m_img  = (const float*)d_in[0];
    const float* psf    = (const float*)d_in[1];
    const float* alpha  = (const float*)d_in[2];
    float* out = (float*)d_out;
    float* ws  = (float*)d_ws;   // needs BATCH*LEN*4 = 128 MB ping buffer

    const int nlayers = in_sizes[2];   // 10
    dim3 grid(LEN / CHUNK, BATCH / ROWS);
    dim3 block(NTH);

    const float* src = nullptr;
    for (int l = 0; l < nlayers; ++l) {
        // choose parity so the final layer writes d_out
        float* dst = (((nlayers - 1 - l) & 1) == 0) ? out : ws;
        rl_layer_kernel<<<grid, block, 0, stream>>>(m_img, psf, alpha,
                                                    src, dst, l, l == 0);
        src = dst;
    }
}